// Siamese_39316130627668
// MI455X (gfx1250) — compile-verified
//
#include <hip/hip_runtime.h>
#include <hip/hip_bf16.h>

#define SS 128
#define QQ 8192
#define HH 230
#define NBLK (QQ / 16)   // 512 gemm blocks

typedef float v2f __attribute__((ext_vector_type(2)));
typedef float v8f __attribute__((ext_vector_type(8)));

// a[s] = sum_h w[h] * x[s,h]^2   (128 rows, one block)
__global__ void row_norm_kernel(const float* __restrict__ x,
                                const float* __restrict__ w,
                                float* __restrict__ arow) {
    int s = threadIdx.x;
    if (s < SS) {
        const float* xr = x + s * HH;
        float acc = 0.f;
        for (int h = 0; h < HH; ++h) {
            float v = xr[h];
            acc = fmaf(w[h], v * v, acc);
        }
        arow[s] = acc;
    }
}

// c[q] = sum_h w[h] * y[q,h]^2   (8192 rows)
__global__ void col_norm_kernel(const float* __restrict__ y,
                                const float* __restrict__ w,
                                float* __restrict__ ccol) {
    int q = blockIdx.x * blockDim.x + threadIdx.x;
    if (q < QQ) {
        const float* yr = y + q * HH;
        float acc = 0.f;
        for (int h = 0; h < HH; ++h) {
            float v = yr[h];
            acc = fmaf(w[h], v * v, acc);
        }
        ccol[q] = acc;
    }
}

// GEMM G = (X .* w) @ Y^T via V_WMMA_F32_16X16X4_F32, fused epilogue:
// logits = -2*G + a[s] + c[q] + bias ; score = sigmoid ; bce accumulated.
__global__ __launch_bounds__(256)
void siamese_wmma_kernel(const float* __restrict__ x, const float* __restrict__ y,
                         const int* __restrict__ label, const float* __restrict__ w,
                         const float* __restrict__ bias,
                         const float* __restrict__ arow, const float* __restrict__ ccol,
                         float* __restrict__ score, float* __restrict__ partial)
{
    __shared__ float red[256];

    const int lane = threadIdx.x & 31;
    const int wave = threadIdx.x >> 5;   // 0..7 -> M tile (wave*16)
    const int half = lane >> 4;          // 0: K=0,1  1: K=2,3
    const int l    = lane & 15;          // A: row M ; B: col N
    const int m0   = wave * 16;
    const int qb   = blockIdx.x * 16;

    // per-lane fragment base pointers (include the K sub-offset for this half)
    const float* xr = x + (m0 + l) * HH + 2 * half;
    const float* yr = y + (qb + l) * HH + 2 * half;
    const float* wr = w + 2 * half;

    v8f acc = {};

    // main K loop: k0 = 0,4,...,224 covers k in [0,228)
    #pragma unroll 4
    for (int k0 = 0; k0 + 4 <= HH; k0 += 4) {
        float2 xv = *(const float2*)(xr + k0);
        float2 wv = *(const float2*)(wr + k0);
        float2 yv = *(const float2*)(yr + k0);
        v2f A; A[0] = xv.x * wv.x; A[1] = xv.y * wv.y;
        v2f B; B[0] = yv.x;        B[1] = yv.y;
        acc = __builtin_amdgcn_wmma_f32_16x16x4_f32(
            /*neg_a=*/false, A, /*neg_b=*/false, B,
            /*c_mod=*/(short)0, acc, /*reuse_a=*/false, /*reuse_b=*/false);
    }

    // tail: k = 228,229 valid (half==0); half==1 slots (k=230,231) are zero-padded
    {
        v2f A = {0.f, 0.f};
        v2f B = {0.f, 0.f};
        if (half == 0) {
            float2 xv = *(const float2*)(xr + 228);
            float2 wv = *(const float2*)(wr + 228);
            float2 yv = *(const float2*)(yr + 228);
            A[0] = xv.x * wv.x; A[1] = xv.y * wv.y;
            B[0] = yv.x;        B[1] = yv.y;
        }
        acc = __builtin_amdgcn_wmma_f32_16x16x4_f32(
            false, A, false, B, (short)0, acc, false, false);
    }

    // epilogue: D layout -> VGPR r : row = m0 + r + 8*half, col = qb + l
    const float bval = *bias;
    const int   q    = qb + l;
    const float cq   = ccol[q];
    float lsum = 0.f;
    #pragma unroll
    for (int r = 0; r < 8; ++r) {
        const int s = m0 + r + 8 * half;
        const float z  = fmaf(-2.f, acc[r], arow[s] + cq + bval);
        const float sc = 1.f / (1.f + expf(-z));
        score[s * QQ + q] = sc;
        const float lab = (float)label[s * QQ + q];
        // stable softplus(z) ; bce = softplus(z) - lab*z
        const float sp = fmaxf(z, 0.f) + log1pf(expf(-fabsf(z)));
        lsum += sp - lab * z;
    }

    // deterministic fixed-order block reduction of BCE
    red[threadIdx.x] = lsum;
    __syncthreads();
    for (int off = 128; off > 0; off >>= 1) {
        if (threadIdx.x < (unsigned)off) red[threadIdx.x] += red[threadIdx.x + off];
        __syncthreads();
    }
    if (threadIdx.x == 0) partial[blockIdx.x] = red[0];
}

// final loss reduction over NBLK=512 partials (deterministic tree)
__global__ __launch_bounds__(256)
void loss_kernel(const float* __restrict__ partial, float* __restrict__ out) {
    __shared__ float red[256];
    red[threadIdx.x] = partial[threadIdx.x] + partial[threadIdx.x + 256];
    __syncthreads();
    for (int off = 128; off > 0; off >>= 1) {
        if (threadIdx.x < (unsigned)off) red[threadIdx.x] += red[threadIdx.x + off];
        __syncthreads();
    }
    if (threadIdx.x == 0) out[SS * QQ] = red[0] * (1.0f / (float)(SS * QQ));
}

extern "C" void kernel_launch(void* const* d_in, const int* in_sizes, int n_in,
                              void* d_out, int out_size, void* d_ws, size_t ws_size,
                              hipStream_t stream) {
    const float* x     = (const float*)d_in[0];   // [S,H]
    const float* y     = (const float*)d_in[1];   // [Q,H]
    const int*   label = (const int*)  d_in[2];   // [S,Q]
    const float* w     = (const float*)d_in[3];   // [H]
    const float* b     = (const float*)d_in[4];   // scalar

    float* out = (float*)d_out;                   // [S*Q] score, then loss at [S*Q]
    float* ws  = (float*)d_ws;
    float* arow    = ws;                          // 128
    float* ccol    = ws + SS;                     // 8192
    float* partial = ws + SS + QQ;                // 512

    row_norm_kernel<<<1, 128, 0, stream>>>(x, w, arow);
    col_norm_kernel<<<QQ / 256, 256, 0, stream>>>(y, w, ccol);
    siamese_wmma_kernel<<<NBLK, 256, 0, stream>>>(x, y, label, w, b,
                                                  arow, ccol, out, partial);
    loss_kernel<<<1, 256, 0, stream>>>(partial, out);
}